// CLAGL_layer_79774722556257
// MI455X (gfx1250) — compile-verified
//
#include <hip/hip_runtime.h>
#include <hip/hip_bf16.h>

// CLAGL layer: 3x COO SpMM (segment_sum of gathered rows * edge vals).
// Memory-bound (AI ~0.24 FLOP/B): optimized for wide coalesced loads +
// hardware global_atomic_add_f32 (output fits in 192MB L2 on MI455X).

#define EMB_D 64

__global__ __launch_bounds__(256) void clagl_zero_f4(float4* __restrict__ out,
                                                     long long n4) {
    long long i = (long long)blockIdx.x * blockDim.x + threadIdx.x;
    if (i < n4) out[i] = float4{0.f, 0.f, 0.f, 0.f};
}

// One edge per 16 lanes; each lane handles one float4 chunk of the D=64 row.
// src row = (col < split) ? srcA[col] : srcB[col - split]   (concat gather)
// dst row = dst + row*64                                     (contig scatter)
__global__ __launch_bounds__(256) void clagl_spmm_scatter(
    const int*   __restrict__ rows,
    const int*   __restrict__ cols,
    const float* __restrict__ vals,
    const float* __restrict__ srcA,
    const float* __restrict__ srcB,
    int split,
    float* __restrict__ dst,
    int nE) {
    long long t = (long long)blockIdx.x * blockDim.x + threadIdx.x;
    int e = (int)(t >> 4);          // edge index
    int c = (int)(t & 15);          // float4 chunk within the 64-float row
    if (e >= nE) return;

    int   row = rows[e];
    int   col = cols[e];
    float v   = vals[e];

    const float* src = (col < split)
        ? (srcA + (size_t)col * EMB_D)
        : (srcB + (size_t)(col - split) * EMB_D);

    float4 x = *(const float4*)(src + c * 4);   // global_load_b128, coalesced

    float* d = dst + (size_t)row * EMB_D + c * 4;
    unsafeAtomicAdd(d + 0, x.x * v);            // global_atomic_add_f32
    unsafeAtomicAdd(d + 1, x.y * v);
    unsafeAtomicAdd(d + 2, x.z * v);
    unsafeAtomicAdd(d + 3, x.w * v);
}

extern "C" void kernel_launch(void* const* d_in, const int* in_sizes, int n_in,
                              void* d_out, int out_size, void* d_ws, size_t ws_size,
                              hipStream_t stream) {
    const float* user_emb  = (const float*)d_in[0];
    const float* item_emb  = (const float*)d_in[1];
    const int*   g_rows    = (const int*)  d_in[2];
    const int*   g_cols    = (const int*)  d_in[3];
    const float* g_vals    = (const float*)d_in[4];
    const int*   u_rows    = (const int*)  d_in[5];
    const int*   u_cols    = (const int*)  d_in[6];
    const float* u_vals    = (const float*)d_in[7];
    const int*   i_rows    = (const int*)  d_in[8];
    const int*   i_cols    = (const int*)  d_in[9];
    const float* i_vals    = (const float*)d_in[10];

    float* out = (float*)d_out;

    const int U = in_sizes[0] / EMB_D;   // 100000
    const int I = in_sizes[1] / EMB_D;   // 50000
    const int N = U + I;                 // 150000
    const int E_G = in_sizes[2];
    const int E_U = in_sizes[5];
    const int E_I = in_sizes[8];

    // --- 1) zero the output accumulators (harness poisons d_out) ---
    {
        long long n4 = (long long)out_size / 4;  // out_size = (N+U+I)*64, /4 exact
        int threads = 256;
        long long blocks = (n4 + threads - 1) / threads;
        clagl_zero_f4<<<(unsigned)blocks, threads, 0, stream>>>((float4*)out, n4);
    }

    // --- 2) graph SpMM over concat(user,item) -> out[0 : N*64) ---
    {
        long long tcount = (long long)E_G * 16;
        long long blocks = (tcount + 255) / 256;
        clagl_spmm_scatter<<<(unsigned)blocks, 256, 0, stream>>>(
            g_rows, g_cols, g_vals, user_emb, item_emb, U,
            out, E_G);
    }

    // --- 3) user SpMM -> out[N*64 : (N+U)*64) ---
    {
        long long tcount = (long long)E_U * 16;
        long long blocks = (tcount + 255) / 256;
        clagl_spmm_scatter<<<(unsigned)blocks, 256, 0, stream>>>(
            u_rows, u_cols, u_vals, user_emb, user_emb, U,
            out + (size_t)N * EMB_D, E_U);
    }

    // --- 4) item SpMM -> out[(N+U)*64 : (N+U+I)*64) ---
    {
        long long tcount = (long long)E_I * 16;
        long long blocks = (tcount + 255) / 256;
        clagl_spmm_scatter<<<(unsigned)blocks, 256, 0, stream>>>(
            i_rows, i_cols, i_vals, item_emb, item_emb, I,
            out + (size_t)(N + U) * EMB_D, E_I);
    }
}